// KKTLoss_46299747451217
// MI455X (gfx1250) — compile-verified
//
#include <hip/hip_runtime.h>
#include <math.h>

// ---------------------------------------------------------------------------
// KKT loss forward for MI455X (gfx1250, wave32, WMMA).
//
// Five TN GEMMs (C = A @ W^T, A:(256xK) row-major, W:(NxK) row-major) feed a
// fused elementwise/reduction epilogue. fp32 WMMA (v_wmma_f32_16x16x4_f32):
// the 432MB of weights stream from HBM exactly once (18.5us floor at
// 23.3TB/s); per-call bf16 conversion would add traffic. Weights are loaded
// non-temporally so they do not evict L2-resident activations.
// ---------------------------------------------------------------------------

typedef float v2f __attribute__((ext_vector_type(2)));
typedef float v8f __attribute__((ext_vector_type(8)));

// Problem constants (fixed by setup_inputs; grid dims must be host-known).
#define BATCH   256
#define NBUS    2000
#define NBUS2   4000      // 2*n_bus
#define NLINE   3000
#define NLINE2  6000      // 2*n_line
#define NGBUS   500
#define NYC     2016      // padded column count for YV/YcV (need cols 0..2001)

// Load one K-chunk (16 k-values = 4 WMMA steps) into fragment buffer s.
//  - B fragments: streamed weights -> non-temporal loads
//  - A fragments: L2-resident activations -> regular loads
#define LOAD_CHUNK(s, koff)                                                   \
    _Pragma("unroll")                                                         \
    for (int kk = 0; kk < 4; ++kk) {                                          \
        bA[s][kk] = __builtin_nontemporal_load(                               \
            (const v2f*)(pb + (koff) + kk * 4));                              \
        _Pragma("unroll")                                                     \
        for (int f = 0; f < 4; ++f)                                           \
            aA[s][f][kk] = *(const v2f*)(pa[f] + (koff) + kk * 4);            \
    }

// Issue the 16 WMMAs consuming fragment buffer s.
#define MMA_CHUNK(s)                                                          \
    _Pragma("unroll")                                                         \
    for (int kk = 0; kk < 4; ++kk) {                                          \
        _Pragma("unroll")                                                     \
        for (int f = 0; f < 4; ++f)                                           \
            acc[f] = __builtin_amdgcn_wmma_f32_16x16x4_f32(                   \
                false, aA[s][f][kk], false, bA[s][kk],                        \
                (short)0, acc[f], false, false);                              \
    }

// ---------------------------------------------------------------------------
// GEMM: C[m, n0+r] = sum_k A[m,k] * W[n0+r, k]     (TN, both K-contiguous)
//   block = 4 waves; wave w covers M rows [w*64, w*64+64) as 4 WMMA tiles;
//   block covers N columns [blockIdx.x*16, +16) and ALL 256 rows, so each
//   weight element is pulled from HBM by exactly one workgroup.
//
// WMMA fragment addressing (ISA 7.12.2), lane L: r = L&15, h = L>>4:
//   A 16x4 f32:  lane holds A[m0+r][k0 + 2h .. 2h+1]      -> one float2
//   B  4x16 f32: lane holds W[n0+r][k0 + 2h .. 2h+1]      -> one float2
//   C 16x16 f32: vgpr i -> C[m0 + i + 8h][n0 + r]
// Inner loop is software-pipelined: ping-pong fragment buffers so the 16
// WMMAs of chunk c overlap the 20 global_load_b64 of chunk c+1.
// ---------------------------------------------------------------------------
__global__ __launch_bounds__(128)
void kkt_gemm_tn_wmma(const float* __restrict__ A,
                      const float* __restrict__ W,
                      float* __restrict__ C,
                      int K, int ldc)
{
    const int n0   = blockIdx.x * 16;
    const int wave = threadIdx.x >> 5;
    const int lane = threadIdx.x & 31;
    const int r    = lane & 15;
    const int h    = lane >> 4;
    const int m0   = wave * 64;

    const float* pa[4];
#pragma unroll
    for (int f = 0; f < 4; ++f)
        pa[f] = A + (long)(m0 + f * 16 + r) * K + 2 * h;
    const float* pb = W + (long)(n0 + r) * K + 2 * h;

    v8f acc[4];
    acc[0] = (v8f){}; acc[1] = (v8f){}; acc[2] = (v8f){}; acc[3] = (v8f){};

    v2f bA[2][4];        // ping-pong B fragments
    v2f aA[2][4][4];     // ping-pong A fragments [buf][mtile][kstep]

    // Pipelined main loop over K chunks of 16 (K % 16 == 0 at all call sites).
    LOAD_CHUNK(0, 0);
    int k = 16;
    for (; k + 32 <= K; k += 32) {
        LOAD_CHUNK(1, k);
        MMA_CHUNK(0);
        LOAD_CHUNK(0, k + 16);
        MMA_CHUNK(1);
    }
    if (k < K) {                 // odd remaining chunk (e.g. K=4000)
        LOAD_CHUNK(1, k);
        MMA_CHUNK(0);
        MMA_CHUNK(1);
    } else {                     // even chunk count consumed (e.g. K=6000)
        MMA_CHUNK(0);
    }

#pragma unroll
    for (int f = 0; f < 4; ++f)
#pragma unroll
        for (int i = 0; i < 8; ++i)
            C[(long)(m0 + f * 16 + i + 8 * h) * ldc + n0 + r] = acc[f][i];
}

// ---------------------------------------------------------------------------
// Fused epilogue: one block per batch row, deterministic LDS tree reduction.
// ---------------------------------------------------------------------------
__global__ __launch_bounds__(256)
void kkt_epilogue(const float* __restrict__ Volt,
                  const float* __restrict__ PQ_Gens,
                  const float* __restrict__ PQ_Loads,
                  const float* __restrict__ mu_g_u,
                  const float* __restrict__ mu_g_d,
                  const float* __restrict__ mu_v_u,
                  const float* __restrict__ mu_v_d,
                  const float* __restrict__ mu_i_u,
                  const float* __restrict__ Gen_max,
                  const float* __restrict__ Gen_min,
                  const float* __restrict__ V_max,
                  const float* __restrict__ V_min,
                  const float* __restrict__ L_limit,
                  const float* __restrict__ C_Pg,
                  const float* __restrict__ C_Qg,
                  const float* __restrict__ Lg_Max,
                  const float* __restrict__ C1,    // Volt @ Y.T      (ld NYC)
                  const float* __restrict__ C2,    // Volt @ Yconj.T  (ld NYC)
                  const float* __restrict__ Ibr,   // (ld NLINE2)
                  const float* __restrict__ Lmat,  // n_o_l_p @ Map_g.T (ld NBUS2)
                  const int*   __restrict__ slack_ptr,
                  float* __restrict__ out)
{
    const int b = blockIdx.x;
    const int t = threadIdx.x;

    const float Lg0 = Lg_Max[0], Lg1 = Lg_Max[1], Lg2 = Lg_Max[2];
    const float inv_ngbus = 1.0f / (float)NGBUS;

    const float* volt = Volt     + (long)b * NBUS2;
    const float* gen  = PQ_Gens  + (long)b * NBUS2;
    const float* load = PQ_Loads + (long)b * NBUS2;
    const float* mgu  = mu_g_u   + (long)b * NBUS2;
    const float* mgd  = mu_g_d   + (long)b * NBUS2;
    const float* mvu  = mu_v_u   + (long)b * NBUS;
    const float* mvd  = mu_v_d   + (long)b * NBUS;
    const float* miu  = mu_i_u   + (long)b * NLINE;
    const float* cpg  = C_Pg     + (long)b * NBUS;
    const float* cqg  = C_Qg     + (long)b * NBUS;
    const float* c1   = C1       + (long)b * NYC;
    const float* c2   = C2       + (long)b * NYC;
    const float* ibr  = Ibr      + (long)b * NLINE2;
    const float* lmat = Lmat     + (long)b * NBUS2;

    float err = 0.0f;
    float sv  = 0.0f;   // sum of Volt[b, n:2n]

    // ---- loop over 2n: power-balance, gen limits, gen duals, dual_term ----
    for (int j = t; j < NBUS2; j += 256) {
        const float g  = gen[j];
        const float gx = Gen_max[j];
        const float gn = Gen_min[j];
        err += load[j] - g;
        const float du = g - gx;
        const float dd = gn - g;
        err += fmaxf(du, 0.0f) + fmaxf(dd, 0.0f);
        const float u = mgu[j];
        const float d = mgd[j];
        err += (fabsf(u * du) + fabsf(d * dd)) * inv_ngbus;
        err += fmaxf(-u, 0.0f) + fmaxf(-d, 0.0f);
        const float cc = (j < NBUS) ? cpg[j] : cqg[j - NBUS];
        err += fabsf(u * Lg1 - d * Lg2 + Lg0 * lmat[j] - cc);
        const float v = volt[j];
        if (j < NBUS) {
            err += v * (c1[j] + c2[j]);   // Volt .* (YV + YcV), low half
        } else {
            sv  += v;                     // for the [n+1] column terms
        }
    }

    // ---- loop over n: voltage-magnitude terms ----
    for (int j = t; j < NBUS; j += 256) {
        const float vr = volt[j];
        const float vi = volt[NBUS + j];
        const float m  = vr * vr + vi * vi;
        const float vx = V_max[j];
        const float vn = V_min[j];
        const float du = m - vx * vx;
        const float dd = vn * vn - m;
        err += fmaxf(du, 0.0f) + fmaxf(dd, 0.0f);
        const float u = mvu[j];
        const float d = mvd[j];
        err += fabsf(u * du) + fabsf(d * dd);
        err += fmaxf(-u, 0.0f) + fmaxf(-d, 0.0f);
    }

    // ---- loop over n_line: branch-current terms ----
    for (int j = t; j < NLINE; j += 256) {
        const float ir = ibr[j];
        const float ii = ibr[NLINE + j];
        const float s  = ir * ir + ii * ii;
        const float ll = L_limit[j];
        const float du = s - ll * ll;
        err += fmaxf(du, 0.0f);
        const float u = miu[j];
        err += fabsf(u * du);
        err += fmaxf(-u, 0.0f);
    }

    // ---- deterministic tree reduction ----
    __shared__ float red_e[256];
    __shared__ float red_s[256];
    red_e[t] = err;
    red_s[t] = sv;
    __syncthreads();
    for (int s = 128; s > 0; s >>= 1) {
        if (t < s) {
            red_e[t] += red_e[t + s];
            red_s[t] += red_s[t + s];
        }
        __syncthreads();
    }
    if (t == 0) {
        float e   = red_e[0];
        float svt = red_s[0];
        e += fabsf(volt[slack_ptr[0]]);                 // |Volt[:, slack]|
        e += svt * (c1[NBUS + 1] + c2[NBUS + 1]);       // sV_hi * (YV+YcV)[n+1]
        out[b] = e;
    }
}

extern "C" void kernel_launch(void* const* d_in, const int* in_sizes, int n_in,
                              void* d_out, int out_size, void* d_ws, size_t ws_size,
                              hipStream_t stream) {
    const float* Volt     = (const float*)d_in[0];
    const float* PQ_Gens  = (const float*)d_in[1];
    const float* PQ_Loads = (const float*)d_in[2];
    const float* n_o_l_p  = (const float*)d_in[3];
    const float* mu_g_u   = (const float*)d_in[4];
    const float* mu_g_d   = (const float*)d_in[5];
    const float* mu_v_u   = (const float*)d_in[6];
    const float* mu_v_d   = (const float*)d_in[7];
    const float* mu_i_u   = (const float*)d_in[8];
    const float* Y        = (const float*)d_in[9];
    const float* Yconj    = (const float*)d_in[10];
    const float* Gen_max  = (const float*)d_in[11];
    const float* Gen_min  = (const float*)d_in[12];
    const float* V_max    = (const float*)d_in[13];
    const float* V_min    = (const float*)d_in[14];
    const float* Ybr      = (const float*)d_in[15];
    const float* IM       = (const float*)d_in[16];
    const float* L_limit  = (const float*)d_in[17];
    const float* C_Pg     = (const float*)d_in[18];
    const float* C_Qg     = (const float*)d_in[19];
    const float* Lg_Max   = (const float*)d_in[20];
    const float* Map_g    = (const float*)d_in[21];
    const int*   slack_p  = (const int*)d_in[25];
    float* out = (float*)d_out;

    // Workspace layout (floats)
    float* C1   = (float*)d_ws;                       // 256 x 2016
    float* C2   = C1  + (long)BATCH * NYC;            // 256 x 2016
    float* T    = C2  + (long)BATCH * NYC;            // 256 x 6000
    float* IbrW = T   + (long)BATCH * NLINE2;         // 256 x 6000
    float* Lmat = IbrW + (long)BATCH * NLINE2;        // 256 x 4000

    dim3 blk(128);
    // YV / YcV: only columns 0..2001 are consumed -> compute 2016 columns.
    kkt_gemm_tn_wmma<<<dim3(NYC / 16), blk, 0, stream>>>(Volt, Y,     C1,   NBUS2,  NYC);
    kkt_gemm_tn_wmma<<<dim3(NYC / 16), blk, 0, stream>>>(Volt, Yconj, C2,   NBUS2,  NYC);
    // T = Volt @ IM.T ; Ibr = T @ Ybr.T
    kkt_gemm_tn_wmma<<<dim3(NLINE2 / 16), blk, 0, stream>>>(Volt, IM,  T,    NBUS2,  NLINE2);
    kkt_gemm_tn_wmma<<<dim3(NLINE2 / 16), blk, 0, stream>>>(T,    Ybr, IbrW, NLINE2, NLINE2);
    // Lmat = n_o_l_p @ Map_g.T  (Lg_Max[0] scaling applied in epilogue)
    kkt_gemm_tn_wmma<<<dim3(NBUS2 / 16), blk, 0, stream>>>(n_o_l_p, Map_g, Lmat, NBUS2, NBUS2);

    kkt_epilogue<<<dim3(BATCH), dim3(256), 0, stream>>>(
        Volt, PQ_Gens, PQ_Loads, mu_g_u, mu_g_d, mu_v_u, mu_v_d, mu_i_u,
        Gen_max, Gen_min, V_max, V_min, L_limit, C_Pg, C_Qg, Lg_Max,
        C1, C2, IbrW, Lmat, slack_p, out);
}